// VectorQuantizer_70411693851194
// MI455X (gfx1250) — compile-verified
//
#include <hip/hip_runtime.h>

typedef float v2f __attribute__((ext_vector_type(2)));
typedef float v8f __attribute__((ext_vector_type(8)));

#define K_CB    1024
#define DIMV    64
#define ROWS_B  32           // rows of X per block (2 WMMA row tiles)
#define NWAVE   8            // waves per block (256 threads, wave32)
#define TILES_W 8            // 16-col tiles per wave (8*8*16 = 1024 cols)
#define N_TOT   32768        // 32*32*32
#define ELEMS   2097152.0f   // 32*32*32*64
#define LDS_STRIDE 68        // padded row stride (dwords): bank-conflict-free b64 reads

// Order-preserving float -> uint key (works for negatives too)
__device__ __forceinline__ unsigned fkey(float f) {
  unsigned b = __float_as_uint(f);
  return (b & 0x80000000u) ? ~b : (b | 0x80000000u);
}
__device__ __forceinline__ float funkey(unsigned k) {
  unsigned b = (k & 0x80000000u) ? (k & 0x7fffffffu) : ~k;
  return __uint_as_float(b);
}

__global__ void vq_init(unsigned* __restrict__ minkey, int* __restrict__ active,
                        float* __restrict__ sqsum) {
  int t = blockIdx.x * blockDim.x + threadIdx.x;
  if (t < K_CB) { minkey[t] = 0xFFFFFFFFu; active[t] = 0; }
  if (t == 0) *sqsum = 0.0f;
}

__global__ __launch_bounds__(256) void vq_main(
    const float* __restrict__ X, const float* __restrict__ C,
    float* __restrict__ out, unsigned* __restrict__ minkey,
    int* __restrict__ active, float* __restrict__ sqsum) {
  __shared__ float xtile[ROWS_B * LDS_STRIDE];   // staged X tile (padded rows)
  __shared__ float x2s[ROWS_B];                  // ||x||^2 per row
  __shared__ float colmin[K_CB];                 // per-column min over block rows
  __shared__ float rbestv[NWAVE][ROWS_B];
  __shared__ int   rbesti[NWAVE][ROWS_B];

  const int lane = threadIdx.x & 31;
  const int wave = threadIdx.x >> 5;
  const int hl   = lane >> 4;                    // half-wave: 0 or 1
  const int l16  = lane & 15;
  const int rowBase = blockIdx.x * ROWS_B;

  // ---- stage X tile (32 rows x 64 f32 = 8KB) into LDS via async global->LDS ----
  // 512 B128 chunks; each of the 256 lanes issues 2 (tracked by ASYNCcnt).
  {
    const float* xbase = X + (size_t)rowBase * DIMV;
    #pragma unroll
    for (int c = 0; c < 2; ++c) {
      int chunk = threadIdx.x + c * 256;         // 0..511
      int row = chunk >> 4;                      // 16 B128 chunks per 256B row
      int seg = chunk & 15;
      unsigned dst = (unsigned)(uintptr_t)xtile +
                     (unsigned)(row * LDS_STRIDE * 4 + seg * 16);
      unsigned long long gaddr =
          (unsigned long long)(uintptr_t)(xbase + row * DIMV + seg * 4);
      asm volatile("global_load_async_to_lds_b128 %0, %1, off"
                   :: "v"(dst), "v"(gaddr) : "memory");
    }
    asm volatile("s_wait_asynccnt 0x0" ::: "memory");
  }
  __syncthreads();

  // ---- ||x||^2 per row, once per block ----
  if (threadIdx.x < ROWS_B) {
    const float* r = &xtile[threadIdx.x * LDS_STRIDE];
    float s = 0.0f;
    #pragma unroll
    for (int e = 0; e < DIMV; ++e) s += r[e] * r[e];
    x2s[threadIdx.x] = s;
  }
  __syncthreads();

  // ---- load A fragments for both row tiles from LDS (reused across 8 col tiles)
  // f32 16x4 A layout: lanes 0-15 hold K=0,1 (vgpr0,1); lanes 16-31 hold K=2,3
  v2f a0[16], a1[16];
  {
    const float* r0 = &xtile[(l16)      * LDS_STRIDE];
    const float* r1 = &xtile[(16 + l16) * LDS_STRIDE];
    #pragma unroll
    for (int kb = 0; kb < 16; ++kb) {
      int kk = kb * 4 + hl * 2;
      a0[kb] = *(const v2f*)(r0 + kk);
      a1[kb] = *(const v2f*)(r1 + kk);
    }
  }
  // x2 values needed by this lane under the C/D layout (vgpr j -> row j + hl*8)
  float rx2a[8], rx2b[8];
  #pragma unroll
  for (int j = 0; j < 8; ++j) {
    rx2a[j] = x2s[j + hl * 8];
    rx2b[j] = x2s[16 + j + hl * 8];
  }

  float bv0[8], bv1[8]; int bi0[8], bi1[8];
  #pragma unroll
  for (int j = 0; j < 8; ++j) {
    bv0[j] = 3.4e38f; bi0[j] = 0;
    bv1[j] = 3.4e38f; bi1[j] = 0;
  }

  // ---- sweep this wave's 8 column tiles of the codebook ----
  for (int t = 0; t < TILES_W; ++t) {
    const int colBase = wave * (TILES_W * 16) + t * 16;
    const float* crow = C + (size_t)(colBase + l16) * DIMV;
    v2f bfrag[16];
    float cp = 0.0f;
    #pragma unroll
    for (int kb = 0; kb < 16; ++kb) {
      int kk = kb * 4 + hl * 2;
      v2f b; b.x = crow[kk]; b.y = crow[kk + 1];
      bfrag[kb] = b;
      cp += b.x * b.x + b.y * b.y;
    }
    float c2 = cp + __shfl_xor(cp, 16, 32);      // ||c||^2 of col (colBase+l16)

    // two independent accumulator chains sharing B: interleaved WMMAs
    v8f acc0 = {}, acc1 = {};
    #pragma unroll
    for (int kb = 0; kb < 16; ++kb) {
      acc0 = __builtin_amdgcn_wmma_f32_16x16x4_f32(
          false, a0[kb], false, bfrag[kb], (short)0, acc0, false, false);
      acc1 = __builtin_amdgcn_wmma_f32_16x16x4_f32(
          false, a1[kb], false, bfrag[kb], (short)0, acc1, false, false);
    }

    const int n = colBase + l16;                 // this lane's column
    float cmin = 3.4e38f;
    #pragma unroll
    for (int j = 0; j < 8; ++j) {
      float d0 = rx2a[j] - 2.0f * acc0[j] + c2;
      float d1 = rx2b[j] - 2.0f * acc1[j] + c2;
      if (d0 < bv0[j]) { bv0[j] = d0; bi0[j] = n; }
      if (d1 < bv1[j]) { bv1[j] = d1; bi1[j] = n; }
      cmin = fminf(cmin, fminf(d0, d1));
    }
    cmin = fminf(cmin, __shfl_xor(cmin, 16, 32)); // combine both row halves
    if (hl == 0) colmin[n] = cmin;                // each col owned by one wave
  }

  // ---- row argmin: reduce across the 16 lanes of each half-wave ----
  #pragma unroll
  for (int j = 0; j < 8; ++j) {
    float v0 = bv0[j]; int i0 = bi0[j];
    float v1 = bv1[j]; int i1 = bi1[j];
    #pragma unroll
    for (int m = 8; m >= 1; m >>= 1) {            // xor 8,4,2,1 stays in group
      float ov0 = __shfl_xor(v0, m, 32); int oi0 = __shfl_xor(i0, m, 32);
      float ov1 = __shfl_xor(v1, m, 32); int oi1 = __shfl_xor(i1, m, 32);
      if (ov0 < v0 || (ov0 == v0 && oi0 < i0)) { v0 = ov0; i0 = oi0; }
      if (ov1 < v1 || (ov1 == v1 && oi1 < i1)) { v1 = ov1; i1 = oi1; }
    }
    if (l16 == 0) {
      rbestv[wave][j + hl * 8]      = v0; rbesti[wave][j + hl * 8]      = i0;
      rbestv[wave][16 + j + hl * 8] = v1; rbesti[wave][16 + j + hl * 8] = i1;
    }
  }
  __syncthreads();

  // ---- per-column global min via order-preserving uint atomicMin ----
  for (int c = threadIdx.x; c < K_CB; c += 256)
    atomicMin(&minkey[c], fkey(colmin[c]));

  // ---- finalize block rows: token, histogram, gather, MSE ----
  if (threadIdx.x < ROWS_B) {
    const int r = threadIdx.x;
    float v = rbestv[0][r]; int tok = rbesti[0][r];
    #pragma unroll
    for (int w = 1; w < NWAVE; ++w) {
      float ov = rbestv[w][r]; int oi = rbesti[w][r];
      if (ov < v || (ov == v && oi < tok)) { v = ov; tok = oi; }
    }
    atomicAdd(&active[tok], 1);
    const float* cv = C + (size_t)tok * DIMV;
    const float* xv = &xtile[r * LDS_STRIDE];
    float*       ov = out + (size_t)(rowBase + r) * DIMV;
    float s = 0.0f;
    #pragma unroll 8
    for (int e = 0; e < DIMV; ++e) {
      float emb = cv[e];
      float dx  = emb - xv[e];
      s += dx * dx;
      ov[e] = emb;                                // straight-through forward
    }
    atomicAdd(sqsum, s);
  }
}

__global__ void vq_final(const unsigned* __restrict__ minkey,
                         const int* __restrict__ active,
                         const float* __restrict__ sqsum,
                         float* __restrict__ loss_out) {
  __shared__ float red[256];
  float s = 0.0f;
  for (int k = threadIdx.x; k < K_CB; k += 256)
    if (active[k] == 0) s += funkey(minkey[k]);
  red[threadIdx.x] = s;
  __syncthreads();
  for (int off = 128; off >= 1; off >>= 1) {
    if ((int)threadIdx.x < off) red[threadIdx.x] += red[threadIdx.x + off];
    __syncthreads();
  }
  if (threadIdx.x == 0) {
    float mse = *sqsum / ELEMS;                   // emb(1.0) + commit(0.25)
    float ent = red[0] / (float)K_CB;
    *loss_out = 1.25f * mse + 0.01f * ent;
  }
}

extern "C" void kernel_launch(void* const* d_in, const int* in_sizes, int n_in,
                              void* d_out, int out_size, void* d_ws, size_t ws_size,
                              hipStream_t stream) {
  (void)in_sizes; (void)n_in; (void)out_size; (void)ws_size;
  const float* X = (const float*)d_in[0];        // [32,32,32,64]
  const float* C = (const float*)d_in[1];        // [1024,64]
  float* out  = (float*)d_out;                   // [2097152] out + [1] loss
  float* loss = out + 2097152;

  unsigned* minkey = (unsigned*)d_ws;            // 1024 u32
  int*      active = (int*)(minkey + K_CB);      // 1024 i32
  float*    sqsum  = (float*)(active + K_CB);    // 1 f32

  vq_init<<<4, 256, 0, stream>>>(minkey, active, sqsum);
  vq_main<<<N_TOT / ROWS_B, 256, 0, stream>>>(X, C, out, minkey, active, sqsum);
  vq_final<<<1, 256, 0, stream>>>(minkey, active, sqsum, loss);
}